// AFTSimpleAttention_15564961481421
// MI455X (gfx1250) — compile-verified
//
#include <hip/hip_runtime.h>
#include <hip/hip_bf16.h>
#include <stdint.h>

// ---------------------------------------------------------------------------
// AFT simple attention (causal) on MI455X / gfx1250.
//
// Pure scan workload, no matmul -> bandwidth bound.
//   min traffic = K x2 + V + Q + out = 5 * 134 MB = 670 MB -> ~29 us @ 23.3 TB/s
//
// k_stats : per (n,h,chunk) online-softmax stats (m, sum_exp) per e    [K read]
// k_carry : tiny cross-chunk combine -> global M + exclusive exp prefix P
// k_main  : full Tensor-Data-Mover pipeline per (n,h,chunk) block:
//             issue TDM K into LDS at byte 0, TDM V into LDS at byte 128K      (cnt=2)
//             s_wait_tensorcnt 1  -> Phase A: exp scan on K (V DMA in flight)
//             s_wait_tensorcnt 0  -> Phase C: w*V prefix from LDS
//             issue TDM Q into LDS at byte 128K (the V buffer) (overlaps the carry-chain spin)
//             chained cross-chunk acc carry (agent-scope acquire/release)
//             s_wait_tensorcnt 0  -> Phase E: out tile in place in the K buffer at LDS byte 0
//             tensor_store_from_lds drains the out tile.
// ---------------------------------------------------------------------------

#define Nn    4
#define Ll    8192
#define Hh    16
#define Ee    64
#define ROWS  512                 // rows (l) per chunk
#define CCH   (Ll / ROWS)         // 16 chunks per (n,h)
#define NH    (Nn * Hh)           // 64 groups
#define NBLK  (NH * CCH)          // 1024 heavy blocks
#define RSTR  (Hh * Ee)           // 1024 floats between consecutive l
#define SEG   (ROWS / 4)          // 128 rows per thread segment

// workspace layout (in floats; flags are ints at FLAG_OFF)
#define ML_OFF   0
#define SL_OFF   (ML_OFF + NBLK * Ee)
#define M_OFF    (SL_OFF + NBLK * Ee)
#define P_OFF    (M_OFF + NH * Ee)
#define ACC_OFF  (P_OFF + NBLK * Ee)
#define FLAG_OFF (ACC_OFF + NBLK * Ee)

// dynamic LDS layout (floats) for k_main
#define LDS_KBUF_F  0                           // [ROWS][Ee]: K -> exp -> prefix -> out
#define LDS_VBUF_F  (ROWS * Ee)                 // [ROWS][Ee]: V, then Q
#define LDS_MASK_F  (2 * ROWS * Ee)             // [ROWS]
#define LDS_SCR_F   (LDS_MASK_F + ROWS)         // [4][64] segment totals
#define LDS_CAR_F   (LDS_SCR_F + 256)           // [64] chunk-carry broadcast
#define SMEM_FLOATS (LDS_CAR_F + 64)            // 66368 floats = 259.25 KB
#define SMEM_BYTES  (SMEM_FLOATS * 4)

// --- TDM availability probes (stderr tells us which path was compiled) -----
#if defined(__HIP_DEVICE_COMPILE__) && __has_builtin(__builtin_amdgcn_tensor_load_to_lds)
#define AFT_USE_TDM 1
#warning "AFT probe: __builtin_amdgcn_tensor_load_to_lds AVAILABLE (TDM load path compiled)"
#else
#define AFT_USE_TDM 0
#if defined(__HIP_DEVICE_COMPILE__)
#warning "AFT probe: __builtin_amdgcn_tensor_load_to_lds MISSING (fallback copy path compiled)"
#endif
#endif

#if defined(__HIP_DEVICE_COMPILE__) && __has_builtin(__builtin_amdgcn_tensor_store_from_lds)
#define AFT_USE_TDM_ST 1
#warning "AFT probe: __builtin_amdgcn_tensor_store_from_lds AVAILABLE (TDM store path compiled)"
#else
#define AFT_USE_TDM_ST 0
#if defined(__HIP_DEVICE_COMPILE__)
#warning "AFT probe: __builtin_amdgcn_tensor_store_from_lds MISSING (direct store path compiled)"
#endif
#endif

typedef unsigned int v4u __attribute__((ext_vector_type(4)));
typedef int          v8i __attribute__((ext_vector_type(8)));
typedef int          v4i __attribute__((ext_vector_type(4)));

// 2D-tile descriptor: 512x64 f32 tile, row stride H*E floats, LDS byte offset.
#if AFT_USE_TDM || AFT_USE_TDM_ST
__device__ __forceinline__ void aft_tdm_desc(const void* gptr, unsigned ldsOff,
                                             v4u* g0o, v8i* g1o) {
    unsigned long long ga = (unsigned long long)(uintptr_t)gptr;
    v4u g0;
    g0.x = 1u;                                      // count=1, user descriptor
    g0.y = ldsOff;                                  // lds_addr (bytes)
    g0.z = (unsigned)(ga & 0xffffffffull);          // global_addr[31:0]
    g0.w = (unsigned)((ga >> 32) & 0x1ffffffull)    // global_addr[56:32]
         | (2u << 30);                              // type = 2 ("image")
    v8i g1;
    g1[0] = (int)(2u << 16);                        // data_size = 4B
    g1[1] = (int)((unsigned)Ee << 16);              // tensor_dim0 = 64
    g1[2] = (int)((unsigned)ROWS << 16);            // tensor_dim1 = 512
    g1[3] = (int)((unsigned)Ee << 16);              // tile_dim0 = 64
    g1[4] = (int)(unsigned)ROWS;                    // tile_dim1 = 512
    g1[5] = (int)(unsigned)RSTR;                    // tensor_dim0_stride = 1024
    g1[6] = 0;
    g1[7] = 0;
    *g0o = g0;
    *g1o = g1;
}
#endif

#if AFT_USE_TDM
__device__ __forceinline__ void aft_tdm_load(const void* gptr, unsigned ldsOff) {
    v4u g0; v8i g1;
    aft_tdm_desc(gptr, ldsOff, &g0, &g1);
    v4i z4 = {0, 0, 0, 0};
#if __clang_major__ >= 23
    v8i z8 = {0, 0, 0, 0, 0, 0, 0, 0};
    __builtin_amdgcn_tensor_load_to_lds(g0, g1, z4, z4, z8, 0);
#else
    __builtin_amdgcn_tensor_load_to_lds(g0, g1, z4, z4, 0);
#endif
}
#endif

#if AFT_USE_TDM_ST
__device__ __forceinline__ void aft_tdm_store(const void* gptr, unsigned ldsOff) {
    v4u g0; v8i g1;
    aft_tdm_desc(gptr, ldsOff, &g0, &g1);
    v4i z4 = {0, 0, 0, 0};
#if __clang_major__ >= 23
    v8i z8 = {0, 0, 0, 0, 0, 0, 0, 0};
    __builtin_amdgcn_tensor_store_from_lds(g0, g1, z4, z4, z8, 0);
#else
    __builtin_amdgcn_tensor_store_from_lds(g0, g1, z4, z4, 0);
#endif
}
#endif

// ---------------------------------------------------------------------------
// k_main: one block per (n,h,chunk), chunk fastest in blockIdx so chained
// predecessors always have lower blockIdx. 256 threads, ~259 KB dynamic LDS.
__global__ void aft_main(const float* __restrict__ Qp,
                         const float* __restrict__ Kp,
                         const float* __restrict__ Vp,
                         const float* __restrict__ maskAdd,
                         float* __restrict__ ws,
                         float* __restrict__ out) {
    extern __shared__ float smem[];
    float* kbuf   = smem + LDS_KBUF_F;                  // [ROWS][Ee]
    float* vbuf   = smem + LDS_VBUF_F;                  // [ROWS][Ee]
    float* maskS  = smem + LDS_MASK_F;
    float* scr    = smem + LDS_SCR_F;                   // [4][64]
    float* carryL = smem + LDS_CAR_F;

    const int bx = blockIdx.x;          // nh*CCH + c  (c fastest -> chain-safe)
    const int nh = bx / CCH, c = bx % CCH;
    const int n  = nh / Hh,  h = nh % Hh;
    const int tid = threadIdx.x;
    const int e = tid & 63, r = tid >> 6;

    const size_t base = (((size_t)n * Ll + (size_t)c * ROWS) * Hh + h) * Ee;

#if AFT_USE_TDM
    // Kick both big input DMAs immediately: K then V (TENSORcnt completes
    // in order, so cnt<=1 means K is resident while V is still streaming).
    if (tid == 0) {
        aft_tdm_load((const void*)(Kp + base), (unsigned)(LDS_KBUF_F * 4));
        aft_tdm_load((const void*)(Vp + base), (unsigned)(LDS_VBUF_F * 4));
    }
#endif

    float* accInc = ws + ACC_OFF;
    int*   flags  = (int*)(ws + FLAG_OFF);

    for (int j = tid; j < ROWS; j += 256)
        maskS[j] = maskAdd[(size_t)n * Ll + (size_t)c * ROWS + j];

#if AFT_USE_TDM
    __builtin_amdgcn_s_wait_tensorcnt(1);   // K tile resident; V in flight
#else
    for (int i = tid; i < ROWS * Ee; i += 256) {
        const int l = i >> 6, ee = i & 63;
        kbuf[i] = Kp[base + (size_t)l * RSTR + ee];
    }
#endif
    __syncthreads();

    const float Mch = ws[M_OFF + (size_t)nh * Ee + e];
    const float Pe  = ws[P_OFF + (size_t)bx * Ee + e];

    // Phase A: exp(K - M) in place, per-thread segment totals. (V DMA overlaps.)
    float segTot = 0.f;
    for (int i = 0; i < SEG; ++i) {
        const int l = r * SEG + i;
        float ex = __expf(kbuf[l * Ee + e] + maskS[l] - Mch);
        segTot += ex;
        kbuf[l * Ee + e] = ex;
    }
    scr[tid] = segTot;
    __syncthreads();

    // Phase B: per-thread exp carry = global P + earlier segments.
    float sRun = Pe;
    for (int rr = 0; rr < r; ++rr) sRun += scr[rr * 64 + e];
#if AFT_USE_TDM
    __builtin_amdgcn_s_wait_tensorcnt(0);   // V tile resident
#endif
    __syncthreads();   // publishes V; scr reads done before phase D rewrite

    // Phase C: w = ex / cumsum(ex); term prefix (w*V) in place.
    float tacc = 0.f;
    for (int i = 0; i < SEG; ++i) {
        const int l = r * SEG + i;
        float ex = kbuf[l * Ee + e];
        sRun += ex;
        float w = __fdividef(ex, sRun);
#if AFT_USE_TDM
        float v = vbuf[l * Ee + e];
#else
        float v = Vp[base + (size_t)l * RSTR + e];
#endif
        tacc += w * v;
        kbuf[l * Ee + e] = tacc;
    }
    scr[tid] = tacc;
    __syncthreads();

#if AFT_USE_TDM
    // V buffer is dead; stream Q into it. The DMA overlaps the carry chain.
    if (tid == 0) aft_tdm_load((const void*)(Qp + base), (unsigned)(LDS_VBUF_F * 4));
#endif

    // Phase D: in-block segment carry + chained cross-chunk acc carry.
    float carryT = 0.f;
    for (int rr = 0; rr < r; ++rr) carryT += scr[rr * 64 + e];

    if (tid == 0 && c > 0) {
        while (__hip_atomic_load(&flags[bx - 1], __ATOMIC_ACQUIRE,
                                 __HIP_MEMORY_SCOPE_AGENT) == 0) {
#if defined(__HIP_DEVICE_COMPILE__)
            __builtin_amdgcn_s_sleep(1);
#endif
        }
    }
    __syncthreads();

    if (tid < 64) {
        float cIn = 0.f;
        if (c > 0)
            cIn = __hip_atomic_load(&accInc[(size_t)(bx - 1) * Ee + tid],
                                    __ATOMIC_ACQUIRE, __HIP_MEMORY_SCOPE_AGENT);
        float bt = scr[tid] + scr[64 + tid] + scr[128 + tid] + scr[192 + tid];
        __hip_atomic_store(&accInc[(size_t)bx * Ee + tid], cIn + bt,
                           __ATOMIC_RELEASE, __HIP_MEMORY_SCOPE_AGENT);
        carryL[tid] = cIn;
        __threadfence();   // stores acked at device scope before flag flips
    }
    __syncthreads();
    if (tid == 0)
        __hip_atomic_store(&flags[bx], 1, __ATOMIC_RELEASE,
                           __HIP_MEMORY_SCOPE_AGENT);

#if AFT_USE_TDM
    __builtin_amdgcn_s_wait_tensorcnt(0);   // Q tile resident
    __syncthreads();
#endif

    const float accCarry = carryL[e] + carryT;

    // Phase E: out = sigmoid(Q) * (carry + local term prefix).
#if AFT_USE_TDM_ST
    for (int i = 0; i < SEG; ++i) {
        const int l = r * SEG + i;
#if AFT_USE_TDM
        float q = vbuf[l * Ee + e];
#else
        float q = Qp[base + (size_t)l * RSTR + e];
#endif
        float sig = __fdividef(1.f, 1.f + __expf(-q));
        kbuf[l * Ee + e] = sig * (accCarry + kbuf[l * Ee + e]);
    }
    __syncthreads();
    if (tid == 0) aft_tdm_store((const void*)(out + base), (unsigned)(LDS_KBUF_F * 4));
    __builtin_amdgcn_s_wait_tensorcnt(0);
#else
    for (int i = 0; i < SEG; ++i) {
        const int l = r * SEG + i;
        const size_t idx = base + (size_t)l * RSTR + e;
#if AFT_USE_TDM
        float q = vbuf[l * Ee + e];
#else
        float q = Qp[idx];
#endif
        float sig = __fdividef(1.f, 1.f + __expf(-q));
        __builtin_nontemporal_store(sig * (accCarry + kbuf[l * Ee + e]), &out[idx]);
    }
#endif
}

// ---------------------------------------------------------------------------
__global__ void aft_init_flags(int* flags) {
    int i = blockIdx.x * blockDim.x + threadIdx.x;
    if (i < NBLK) flags[i] = 0;
}

// ---------------------------------------------------------------------------
// k_stats: one block per (n,h,chunk). 256 threads = (r in 0..3) x (e in 0..63).
__global__ void aft_stats(const float* __restrict__ K,
                          const float* __restrict__ maskAdd,
                          float* __restrict__ ws) {
    __shared__ float maskS[ROWS];
    __shared__ float cm[256];
    __shared__ float cs[256];

    const int bx = blockIdx.x;          // = nh*CCH + c
    const int nh = bx / CCH, c = bx % CCH;
    const int n  = nh / Hh,  h = nh % Hh;
    const int tid = threadIdx.x;
    const int e = tid & 63, r = tid >> 6;

    for (int j = tid; j < ROWS; j += 256)
        maskS[j] = maskAdd[(size_t)n * Ll + (size_t)c * ROWS + j];
    __syncthreads();

    const size_t base = (((size_t)n * Ll + (size_t)c * ROWS) * Hh + h) * Ee;

    float m = -INFINITY, s = 0.f;
    for (int i = 0; i < SEG; ++i) {
        const int l = r * SEG + i;
        float x = K[base + (size_t)l * RSTR + e] + maskS[l];
        if (x > m) { s = s * __expf(m - x) + 1.f; m = x; }
        else       { s += __expf(x - m); }
    }
    cm[tid] = m;
    cs[tid] = s;
    __syncthreads();

    if (tid < 64) {
        float Mv = cm[tid];
        Mv = fmaxf(Mv, cm[64 + tid]);
        Mv = fmaxf(Mv, cm[128 + tid]);
        Mv = fmaxf(Mv, cm[192 + tid]);
        float Sv = 0.f;
        #pragma unroll
        for (int rr = 0; rr < 4; ++rr) {
            float sv = cs[rr * 64 + tid];
            if (sv > 0.f) Sv += sv * __expf(cm[rr * 64 + tid] - Mv);
        }
        ws[ML_OFF + (size_t)bx * Ee + tid] = Mv;
        ws[SL_OFF + (size_t)bx * Ee + tid] = Sv;
    }
}

// ---------------------------------------------------------------------------
// k_carry: one thread per channel (nh,e). Global max over chunks + exclusive
// prefix of rescaled exp-sums. Tiny (4096 threads, ~0.5 MB traffic).
__global__ void aft_carry(float* __restrict__ ws) {
    const int ch = blockIdx.x * blockDim.x + threadIdx.x;   // 0..NH*Ee-1
    if (ch >= NH * Ee) return;
    const int nh = ch / Ee, e = ch % Ee;

    float mv[CCH], sv[CCH];
    #pragma unroll
    for (int c = 0; c < CCH; ++c) {
        mv[c] = ws[ML_OFF + ((size_t)nh * CCH + c) * Ee + e];
        sv[c] = ws[SL_OFF + ((size_t)nh * CCH + c) * Ee + e];
    }
    float M = mv[0];
    #pragma unroll
    for (int c = 1; c < CCH; ++c) M = fmaxf(M, mv[c]);
    ws[M_OFF + ch] = M;

    float p = 0.f;
    #pragma unroll
    for (int c = 0; c < CCH; ++c) {
        ws[P_OFF + ((size_t)nh * CCH + c) * Ee + e] = p;
        if (sv[c] > 0.f) p += sv[c] * __expf(mv[c] - M);
    }
}

// ---------------------------------------------------------------------------
extern "C" void kernel_launch(void* const* d_in, const int* in_sizes, int n_in,
                              void* d_out, int out_size, void* d_ws, size_t ws_size,
                              hipStream_t stream) {
    const float* Qp  = (const float*)d_in[0];   // queries
    const float* Kp  = (const float*)d_in[1];   // keys
    const float* Vp  = (const float*)d_in[2];   // values
    const float* MAp = (const float*)d_in[3];   // key_lengths_add [N,L]
    float* out = (float*)d_out;
    float* ws  = (float*)d_ws;
    int* flags = (int*)(ws + FLAG_OFF);

    (void)in_sizes; (void)n_in; (void)out_size; (void)ws_size;
    (void)hipFuncSetAttribute((const void*)aft_main,
                              hipFuncAttributeMaxDynamicSharedMemorySize,
                              SMEM_BYTES);

    aft_init_flags<<<(NBLK + 255) / 256, 256, 0, stream>>>(flags);
    aft_stats<<<NBLK, 256, 0, stream>>>(Kp, MAp, ws);
    aft_carry<<<(NH * Ee + 255) / 256, 256, 0, stream>>>(ws);
    aft_main<<<NBLK, 256, SMEM_BYTES, stream>>>(Qp, Kp, Vp, MAp, ws, out);
}